// Model_33182917329026
// MI455X (gfx1250) — compile-verified
//
#include <hip/hip_runtime.h>
#include <math.h>

#define IO  64
#define LIN 1024
#define MEM 2048

typedef __attribute__((ext_vector_type(2))) float v2f;
typedef __attribute__((ext_vector_type(8))) float v8f;

// One wave computes y[r0..r0+15] = bias[r0..r0+15] + W[r0..r0+15, 0:K] . v
// W row-major with row stride ldw (elements); K % 32 == 0; vsh = LDS copy of v.
// V_WMMA_F32_16X16X4_F32 with the vector broadcast across all 16 B columns ->
// every D column holds the correct result (no masking, layout-robust in N).
// Result: lanes with (lane&15)==0 hold rows r0 + (lane>>4)*8 + i in y[i].
__device__ __forceinline__ v8f gemv16_wave(const float* __restrict__ W,
                                           int K, int ldw,
                                           const float* __restrict__ bias, int r0,
                                           const float* vsh) {
  const int lane = threadIdx.x & 31;
  const int col  = lane & 15;   // A: row index within the 16-row tile
  const int half = lane >> 4;   // A: k-pair selector (0 -> k,k+1 ; 1 -> k+2,k+3)
  const float* Wrow = W + (size_t)(r0 + col) * (size_t)ldw + 2 * half;

  v8f acc0 = {0.f, 0.f, 0.f, 0.f, 0.f, 0.f, 0.f, 0.f};
  v8f acc1 = {0.f, 0.f, 0.f, 0.f, 0.f, 0.f, 0.f, 0.f};
  if (bias) {
#pragma unroll
    for (int i = 0; i < 8; ++i) acc0[i] = bias[r0 + half * 8 + i];  // bias in C
  }

  for (int k0 = 0; k0 < K; k0 += 32) {
    __builtin_prefetch(Wrow + k0 + 1024, 0, 3);  // global_prefetch_b8, stream ahead
#pragma unroll
    for (int k = k0; k < k0 + 32; k += 8) {
      v2f a0 = *(const v2f*)(Wrow + k);          // global_load_b64 of this row
      v2f a1 = *(const v2f*)(Wrow + k + 4);
      v2f b0, b1;                                 // B: v broadcast to all columns
      b0.x = vsh[k + 2 * half];
      b0.y = vsh[k + 2 * half + 1];
      b1.x = vsh[k + 4 + 2 * half];
      b1.y = vsh[k + 4 + 2 * half + 1];
      acc0 = __builtin_amdgcn_wmma_f32_16x16x4_f32(false, a0, false, b0,
                                                   (short)0, acc0, false, false);
      acc1 = __builtin_amdgcn_wmma_f32_16x16x4_f32(false, a1, false, b1,
                                                   (short)0, acc1, false, false);
    }
  }
  return acc0 + acc1;
}

__device__ __forceinline__ float sigmoidf_(float x) {
  return 1.f / (1.f + expf(-x));
}

// ---------------------------------------------------------------- stage A
// All work independent of the serial chain, in one wide launch:
//   blocks [0,16):    p   = W_proc @ x    + b_proc   (1024 x 64)
//   blocks [16,112):  gh0 = w_hh0 @ mem0 + b_hh0    (6144 x 2048)
//   blocks [112,208): gh1 = w_hh1 @ mem1 + b_hh1    (6144 x 2048)
// 128 threads = 4 waves, one 16-row tile per wave.
__global__ void k_pre(const float* __restrict__ x,
                      const float* __restrict__ W_proc, const float* __restrict__ b_proc,
                      const float* __restrict__ memory,
                      const float* __restrict__ w_hh0, const float* __restrict__ b_hh0,
                      const float* __restrict__ w_hh1, const float* __restrict__ b_hh1,
                      float* __restrict__ p, float* __restrict__ gh0,
                      float* __restrict__ gh1) {
  __shared__ float sv[MEM];
  const int b = blockIdx.x;
  const float *W, *bias, *v;
  float* dst;
  int K, tile0;
  if (b < 16)       { W = W_proc; bias = b_proc; v = x;            dst = p;   K = IO;  tile0 = b * 4; }
  else if (b < 112) { W = w_hh0;  bias = b_hh0;  v = memory;       dst = gh0; K = MEM; tile0 = (b - 16) * 4; }
  else              { W = w_hh1;  bias = b_hh1;  v = memory + MEM; dst = gh1; K = MEM; tile0 = (b - 112) * 4; }

  for (int i = threadIdx.x; i < K; i += blockDim.x) sv[i] = v[i];
  __syncthreads();

  const int wave = threadIdx.x >> 5;
  const int lane = threadIdx.x & 31;
  const int r0 = (tile0 + wave) * 16;
  v8f y = gemv16_wave(W, K, K, bias, r0, sv);
  if ((lane & 15) == 0) {
    const int base = r0 + (lane >> 4) * 8;
#pragma unroll
    for (int i = 0; i < 8; ++i) dst[base + i] = y[i];
  }
}

// ---------------------------------------------------------------- stage B/C
// gi = w_ih @ vin + b_ih for 16 h-rows (3 gate tiles, one per wave), then the
// fused GRU gate math against precomputed gh.  96 threads = 3 waves.
__global__ void k_gicomb(const float* __restrict__ w_ih, const float* __restrict__ b_ih,
                         const float* __restrict__ vin, int Kin,
                         const float* __restrict__ gh,     // [3*MEM] precomputed
                         const float* __restrict__ hprev,
                         float* __restrict__ hout, float* __restrict__ hmem) {
  __shared__ float sv[MEM];
  __shared__ float sres[3 * 16];
  for (int i = threadIdx.x; i < Kin; i += blockDim.x) sv[i] = vin[i];
  __syncthreads();

  const int wave = threadIdx.x >> 5;   // gate: 0=r 1=z 2=n
  const int lane = threadIdx.x & 31;
  const int j0 = blockIdx.x * 16;

  v8f y = gemv16_wave(w_ih, Kin, Kin, b_ih, wave * MEM + j0, sv);
  if ((lane & 15) == 0) {
    const int base = wave * 16 + (lane >> 4) * 8;
#pragma unroll
    for (int i = 0; i < 8; ++i) sres[base + i] = y[i];
  }
  __syncthreads();

  if (threadIdx.x < 16) {
    const int t = threadIdx.x;
    const int j = j0 + t;
    const float r = sigmoidf_(sres[0 * 16 + t] + gh[0 * MEM + j]);
    const float z = sigmoidf_(sres[1 * 16 + t] + gh[1 * MEM + j]);
    const float n = tanhf(sres[2 * 16 + t] + r * gh[2 * MEM + j]);
    const float h = (1.f - z) * n + z * hprev[j];
    hout[j] = h;   // workspace, feeds next stage
    hmem[j] = h;   // new_memory slice of d_out
  }
}

// ---------------------------------------------------------------- stage D
// K-split output GEMV: block s owns k-slice [s*64, s*64+64).
//   m = h1 + residual (slice), new_residual = sigmoid(m) (slice),
//   part[s][0..63] = W_out[:, slice] @ m[slice]   (no bias)
// 128 threads = 4 waves, 16 rows each.
__global__ void k_outpart(const float* __restrict__ h1, const float* __restrict__ residual,
                          const float* __restrict__ Wo,
                          float* __restrict__ part, float* __restrict__ new_res) {
  __shared__ float sm[64];
  const int s = blockIdx.x;
  const int koff = s * 64;
  if (threadIdx.x < 64) {
    const int idx = koff + threadIdx.x;
    const float v = h1[idx] + residual[idx];
    sm[threadIdx.x] = v;
    new_res[idx] = sigmoidf_(v);
  }
  __syncthreads();

  const int wave = threadIdx.x >> 5;
  const int lane = threadIdx.x & 31;
  const int r0 = wave * 16;
  v8f y = gemv16_wave(Wo + koff, 64, MEM, nullptr, r0, sm);
  if ((lane & 15) == 0) {
    const int base = r0 + (lane >> 4) * 8;
#pragma unroll
    for (int i = 0; i < 8; ++i) part[s * IO + base + i] = y[i];
  }
}

// ---------------------------------------------------------------- stage E
// out[i] = b_out[i] + sum_s part[s][i]   (deterministic reduce, 32 slices)
__global__ void k_outreduce(const float* __restrict__ part, const float* __restrict__ bo,
                            float* __restrict__ out) {
  const int i = threadIdx.x;  // 64 threads
  float acc = bo[i];
  for (int s = 0; s < 32; ++s) acc += part[s * IO + i];
  out[i] = acc;
}

extern "C" void kernel_launch(void* const* d_in, const int* in_sizes, int n_in,
                              void* d_out, int out_size, void* d_ws, size_t ws_size,
                              hipStream_t stream) {
  const float* x        = (const float*)d_in[0];
  const float* residual = (const float*)d_in[1];
  const float* memory   = (const float*)d_in[2];   // [2][1][MEM]
  const float* W_proc   = (const float*)d_in[3];
  const float* b_proc   = (const float*)d_in[4];
  const float* w_ih0    = (const float*)d_in[5];
  const float* w_hh0    = (const float*)d_in[6];
  const float* b_ih0    = (const float*)d_in[7];
  const float* b_hh0    = (const float*)d_in[8];
  const float* w_ih1    = (const float*)d_in[9];
  const float* w_hh1    = (const float*)d_in[10];
  const float* b_ih1    = (const float*)d_in[11];
  const float* b_hh1    = (const float*)d_in[12];
  const float* W_out    = (const float*)d_in[13];
  const float* b_out    = (const float*)d_in[14];

  // d_out: output[64] | new_residual[2048] | new_memory[2*2048]
  float* out      = (float*)d_out;
  float* new_res  = out + IO;
  float* mem0_out = new_res + MEM;
  float* mem1_out = mem0_out + MEM;

  // workspace layout (floats)
  float* p    = (float*)d_ws;       // 1024
  float* h0   = p + LIN;            // 2048
  float* h1   = h0 + MEM;           // 2048
  float* gh0  = h1 + MEM;           // 6144
  float* gh1  = gh0 + 3 * MEM;      // 6144
  float* part = gh1 + 3 * MEM;      // 32*64

  // A: everything independent of the serial GRU chain, one wide launch
  k_pre<<<dim3(208), dim3(128), 0, stream>>>(x, W_proc, b_proc, memory,
                                             w_hh0, b_hh0, w_hh1, b_hh1,
                                             p, gh0, gh1);
  // B: layer-0 input gates + gate math -> h0
  k_gicomb<<<dim3(MEM / 16), dim3(96), 0, stream>>>(w_ih0, b_ih0, p, LIN,
                                                    gh0, memory, h0, mem0_out);
  // C: layer-1 input gates + gate math -> h1
  k_gicomb<<<dim3(MEM / 16), dim3(96), 0, stream>>>(w_ih1, b_ih1, h0, MEM,
                                                    gh1, memory + MEM, h1, mem1_out);
  // D: K-split output GEMV partials + new_residual
  k_outpart<<<dim3(32), dim3(128), 0, stream>>>(h1, residual, W_out, part, new_res);
  // E: deterministic reduction + bias
  k_outreduce<<<dim3(1), dim3(64), 0, stream>>>(part, b_out, out);
}